// ROIAlign_88227218195076
// MI455X (gfx1250) — compile-verified
//
#include <hip/hip_runtime.h>
#include <hip/hip_bf16.h>

// ROIAlign on MI455X (gfx1250).
//
// Roofline: output = 11616*64*49 f32 = 145.7 MB of stores -> ~6.3 us floor at
// 23.3 TB/s HBM. Features are 1.48 MB (L2-resident, 192 MB L2); the 4-tap
// gathers hit L1/L2. Compute is ~4 FMA/output -> trivial. So the kernel is
// store-drain bound, and the CDNA5 feature that matters is the async
// LDS->global store path (ASYNCcnt): compute each ROI's 12.25 KB tile into a
// double-buffered LDS staging area, fire global_store_async_from_lds_b128,
// and overlap the drain with the next ROI's compute. wave32, 8 waves/block.
//
// wmma=0 is intentional: the 4 tap locations vary per sample point (gather,
// not a shared-K contraction). Densifying via separable interpolation would
// inflate FLOPs ~19x and require the slow f32 16x16x4 WMMA path — strictly
// worse than the store floor this pipeline already achieves.

#define HTOT    38
#define WTOT    38
#define CCH     64
#define OS      7
#define NPTS    49                 // 7x7 sample points
#define PLANE   (HTOT * WTOT)      // 1444
#define ROI_OUT (CCH * NPTS)       // 3136 floats  (12.25 KB)
#define CHUNKS  (ROI_OUT / 4)      // 784 b128 chunks
#define BLOCK   256

__device__ __forceinline__ unsigned lds_offset_of(const void* p) {
  // Generic pointers into LDS carry the byte offset in the low 32 bits
  // (flat LDS aperture: addr[31:0] = LDS offset).
  return (unsigned)(unsigned long long)(uintptr_t)p;
}

__device__ __forceinline__ void wait_asynccnt_le3() {
#if __has_builtin(__builtin_amdgcn_s_wait_asynccnt)
  __builtin_amdgcn_s_wait_asynccnt(3);
#else
  asm volatile("s_wait_asynccnt 0x3" ::: "memory");
#endif
}

__device__ __forceinline__ void async_store_b128(const float* gaddr, unsigned lds_byte_off) {
  unsigned long long ga = (unsigned long long)(uintptr_t)gaddr;
  // GV mode: 64-bit per-lane global address in a VGPR pair, LDS byte
  // address in a single VGPR. Tracked with ASYNCcnt.
  asm volatile("global_store_async_from_lds_b128 %0, %1, off"
               :: "v"(ga), "v"(lds_byte_off)
               : "memory");
}

__global__ __launch_bounds__(BLOCK) void roialign_gfx1250_kernel(
    const float* __restrict__ feat,   // (B, 64, 38, 38)
    const float* __restrict__ rois,   // (B, Nb, 4) -> flat (N, 4)
    float* __restrict__ out,          // (N, 64, 7, 7)
    int nroi, int nb_per_batch)
{
  __shared__ alignas(16) float s_buf[2][ROI_OUT];   // 25088 B, double-buffered
  __shared__ alignas(16) int   s_off[NPTS][4];      // 784 B
  __shared__ alignas(16) float s_w[NPTS][4];        // 784 B

  const int tid = threadIdx.x;
  int it = 0;

  for (int n = blockIdx.x; n < nroi; n += gridDim.x, ++it) {
    const int buf = it & 1;

    // ---- Phase 1: per-sample-point taps (49 threads) ----
    // Safe to run while previous iteration's async stores (which only read
    // s_buf) are still draining: previous phase-2->phase-3 barrier already
    // ordered all s_off/s_w readers before this overwrite.
    if (tid < NPTS) {
      const float4 r = *(const float4*)(rois + (size_t)n * 4);
      const float x1 = r.x, y1 = r.y, x2 = r.z, y2 = r.w;
      const float w = fmaxf(x2 - x1, 1.0f);
      const float h = fmaxf(y2 - y1, 1.0f);
      const int px = tid % OS, py = tid / OS;
      const float x = x1 + ((float)px / 6.0f) * w;
      const float y = y1 + ((float)py / 6.0f) * h;
      const float x0 = floorf(x), y0 = floorf(y);

      float wx[2], wy[2], vx[2], vy[2];
      int   xi[2], yi[2];
#pragma unroll
      for (int d = 0; d < 2; ++d) {
        const float xc = x0 + (float)d;
        const float yc = y0 + (float)d;
        wx[d] = 1.0f - fabsf(x - xc);
        wy[d] = 1.0f - fabsf(y - yc);
        vx[d] = (xc >= 0.0f && xc <= (float)(WTOT - 1)) ? 1.0f : 0.0f;
        vy[d] = (yc >= 0.0f && yc <= (float)(HTOT - 1)) ? 1.0f : 0.0f;
        int xii = (int)xc; xii = xii < 0 ? 0 : (xii > WTOT - 1 ? WTOT - 1 : xii);
        int yii = (int)yc; yii = yii < 0 ? 0 : (yii > HTOT - 1 ? HTOT - 1 : yii);
        xi[d] = xii; yi[d] = yii;
      }
#pragma unroll
      for (int dy = 0; dy < 2; ++dy)
#pragma unroll
        for (int dx = 0; dx < 2; ++dx) {
          const int k = dy * 2 + dx;
          s_off[tid][k] = yi[dy] * WTOT + xi[dx];
          s_w[tid][k]   = wx[dx] * wy[dy] * vx[dx] * vy[dy];
        }
    }

    // Buffer-reuse guard: async stores reading s_buf[buf] were issued two
    // iterations ago. In-order async completion: own-wave outstanding <= 3
    // means only the most recent phase's tail remains. The barrier makes the
    // guarantee block-wide before anyone rewrites s_buf[buf], and also
    // publishes the s_off/s_w writes from phase 1.
    wait_asynccnt_le3();
    __syncthreads();

    // ---- Phase 2: gather + weighted sum into LDS tile ----
    // j walks the ROI tile in output order (j = c*49 + p). Stepping by 256
    // advances (c,p) by (+5,+11) mod 49 — track incrementally, no division.
    const float* fbase = feat + (size_t)(n / nb_per_batch) * (size_t)(CCH * PLANE);
    {
      int c = tid / NPTS;                // <= 5, one magic-mul division
      int p = tid - c * NPTS;
      const float* fc = fbase + c * PLANE;
#pragma unroll 4
      for (int j = tid; j < ROI_OUT; j += BLOCK) {
        const int4   o4 = *(const int4*)  s_off[p];
        const float4 w4 = *(const float4*) s_w[p];
        float acc;
        acc = w4.x * fc[o4.x];
        acc = fmaf(w4.y, fc[o4.y], acc);
        acc = fmaf(w4.z, fc[o4.z], acc);
        acc = fmaf(w4.w, fc[o4.w], acc);
        s_buf[buf][j] = acc;
        p  += 11;
        fc += 5 * PLANE;
        if (p >= NPTS) { p -= NPTS; fc += PLANE; }
      }
    }
    __syncthreads();   // LDS tile complete (DScnt drained) before async reads

    // ---- Phase 3: fire-and-forget async LDS -> global stores ----
    float* gbase = out + (size_t)n * ROI_OUT;
    const unsigned lbase = lds_offset_of(&s_buf[buf][0]);
    for (int j = tid; j < CHUNKS; j += BLOCK) {
      async_store_b128(gbase + j * 4, lbase + (unsigned)(j * 16));
    }
    // No wait here: next iteration's phase 1 + gather overlap the drain.
    // s_endpgm performs an implicit wait-idle, so all data lands.
  }
}

extern "C" void kernel_launch(void* const* d_in, const int* in_sizes, int n_in,
                              void* d_out, int out_size, void* d_ws, size_t ws_size,
                              hipStream_t stream) {
  const float* feat = (const float*)d_in[0];   // (4, 64, 38, 38) f32
  const float* rois = (const float*)d_in[1];   // (4, 2904, 4)   f32
  float* out = (float*)d_out;                  // (N, 64, 7, 7)  f32

  const int nroi = in_sizes[1] / 4;                        // 11616
  const int batches = in_sizes[0] / (CCH * PLANE);         // 4
  const int nb_per_batch = nroi / (batches > 0 ? batches : 1); // 2904

  // Persistent blocks: ~8 ROIs each so the double-buffered async-store
  // pipeline reaches steady state; 1452 blocks x 8 waves saturates the WGPs.
  int blocks = (nroi + 7) / 8;
  if (blocks < 1) blocks = 1;

  roialign_gfx1250_kernel<<<blocks, BLOCK, 0, stream>>>(feat, rois, out,
                                                        nroi, nb_per_batch);
}